// GCN_63625645523608
// MI455X (gfx1250) — compile-verified
//
#include <hip/hip_runtime.h>

// GCN 2-layer for MI455X (gfx1250, wave32).
// Design notes:
//  * GEMMs use native fp32 WMMA (V_WMMA_F32_16X16X4_F32) -> exact fp32 math
//    on the matrix pipe.
//  * B matrix is staged in LDS in *fragment order* so each WMMA B operand is
//    one aligned ds_load_b64 (no v_mov shuffles in the inner loop).
//  * All intermediates (h1/agg1 25.6MB, g2 6.4MB, dinv 0.4MB) are L2-resident
//    (192MB L2), so the random edge gather/scatter stays on-chip; HBM traffic
//    is only x (102MB) + edges (13MB) + out (6.4MB) ~ 5-6us at 23.3TB/s.
//  * Scatter-add uses global_atomic_add_f32; work set is identical every call.

typedef float v2f __attribute__((ext_vector_type(2)));
typedef float v8f __attribute__((ext_vector_type(8)));

// ---------------------------------------------------------------------------
// WMMA fp32 GEMM: C[M x N] = A[M x K] * B[K x N], row-major.
// B staged in LDS, permuted to fragment order (K*N*4 bytes; 64KB / 4KB).
// Each wave computes a 16-row strip across all N columns (N/16 wmma tiles).
//
// Fragment layouts per CDNA5 ISA 7.12.2:
//  A (16x4 f32): lanes 0-15 hold row M=lane, K={k,k+1} in v0,v1;
//                lanes 16-31 hold row M=lane-16, K={k+2,k+3}.
//  B (4x16 f32): v0: lanes0-15 = row k+0 (col=lane), lanes16-31 = row k+2;
//                v1: lanes0-15 = row k+1,            lanes16-31 = row k+3.
//  C/D (16x16): VGPR r: lanes0-15 -> M=r, lanes16-31 -> M=r+8, col=lane&15.
//
// LDS swizzle: lB[((k>>2)*N + n)*4 + (k&3)] = B[k*N + n], so the pair
// {B[kb*4+2*hi][n], B[kb*4+2*hi+1][n]} is 8B-contiguous per lane.
// ---------------------------------------------------------------------------
template <int K, int N>
__global__ __launch_bounds__(256) void wmma_gemm_f32(
    const float* __restrict__ A, const float* __restrict__ B,
    float* __restrict__ C, int M) {
  __shared__ float lB[K * N];
  for (int i = threadIdx.x; i < K * N; i += 256) {
    int k = i / N, n = i % N;
    lB[((k >> 2) * N + n) * 4 + (k & 3)] = B[i];
  }
  __syncthreads();

  const int lane  = threadIdx.x & 31;
  const int wave  = threadIdx.x >> 5;
  const int mbase = (blockIdx.x * 8 + wave) * 16;
  if (mbase >= M) return;  // wave-uniform: EXEC stays all-ones for WMMA

  const int mr = lane & 15;   // row within 16-tile (A) / column (B, C)
  const int hi = lane >> 4;   // half-wave select

  const float* arow = A + (size_t)(mbase + mr) * K + 2 * hi;
  // Per-lane base into swizzled B: fragment floats at
  //   lB[kb*N*4 + (t*16 + mr)*4 + 2*hi + {0,1}]
  const float* brow = lB + mr * 4 + 2 * hi;

  constexpr int NT = N / 16;
  v8f acc[NT];
  v8f zero = {};
#pragma unroll
  for (int t = 0; t < NT; ++t) acc[t] = zero;

  for (int kb = 0; kb < K / 4; ++kb) {
    v2f a = *(const v2f*)(arow + kb * 4);  // 8B-aligned global_load_b64
#pragma unroll
    for (int t = 0; t < NT; ++t) {
      v2f b = *(const v2f*)(brow + kb * (N * 4) + t * 64);  // ds_load_b64
      acc[t] = __builtin_amdgcn_wmma_f32_16x16x4_f32(
          /*neg_a=*/false, a, /*neg_b=*/false, b,
          /*c_mod=*/(short)0, acc[t], /*reuse_a=*/false, /*reuse_b=*/false);
    }
  }

#pragma unroll
  for (int t = 0; t < NT; ++t)
#pragma unroll
    for (int r = 0; r < 8; ++r)
      C[(size_t)(mbase + 8 * hi + r) * N + t * 16 + mr] = acc[t][r];
}

// ---------------------------------------------------------------------------
// Degree / normalization
// ---------------------------------------------------------------------------
__global__ void k_deg_init(float* __restrict__ deg, int n) {
  int i = blockIdx.x * blockDim.x + threadIdx.x;
  if (i < n) deg[i] = 1.0f;  // self-loop contributes 1 to every in-degree
}

__global__ void k_deg_count(const int* __restrict__ dst,
                            float* __restrict__ deg, int e) {
  int i = blockIdx.x * blockDim.x + threadIdx.x;
  if (i < e) atomicAdd(&deg[dst[i]], 1.0f);
}

__global__ void k_rsqrt_inplace(float* __restrict__ d, int n) {
  int i = blockIdx.x * blockDim.x + threadIdx.x;
  if (i < n) d[i] = rsqrtf(d[i]);  // deg >= 1 always (self-loops)
}

// ---------------------------------------------------------------------------
// Aggregation: agg[i] = h[i]*dinv[i]^2 (self-loop term), then edge scatter
// agg[dst] += h[src]*dinv[src]*dinv[dst]. F/4 threads per row/edge, float4.
// ---------------------------------------------------------------------------
template <int F>
__global__ void k_self_init(const float* __restrict__ h,
                            const float* __restrict__ dinv,
                            float* __restrict__ agg, int n) {
  constexpr int PER = F / 4;
  int t = blockIdx.x * blockDim.x + threadIdx.x;
  if (t >= n * PER) return;
  int node = t / PER;
  float di = dinv[node];
  float s = di * di;
  float4 v = ((const float4*)h)[t];
  float4 o = {v.x * s, v.y * s, v.z * s, v.w * s};
  ((float4*)agg)[t] = o;
}

template <int F>
__global__ __launch_bounds__(256) void k_scatter(
    const int* __restrict__ src, const int* __restrict__ dst,
    const float* __restrict__ dinv, const float* __restrict__ h,
    float* __restrict__ agg, int e) {
  constexpr int PER = F / 4;
  int t = blockIdx.x * blockDim.x + threadIdx.x;
  int ed = t / PER;
  int q  = t % PER;
  if (ed >= e) return;
  int s = src[ed], d = dst[ed];
  float nrm = dinv[s] * dinv[d];
  float4 v = ((const float4*)(h + (size_t)s * F))[q];  // L2-resident gather
  float* ap = agg + (size_t)d * F + q * 4;
  atomicAdd(ap + 0, v.x * nrm);
  atomicAdd(ap + 1, v.y * nrm);
  atomicAdd(ap + 2, v.z * nrm);
  atomicAdd(ap + 3, v.w * nrm);
}

template <int F>
__global__ void k_bias_relu(float* __restrict__ agg,
                            const float* __restrict__ b, int n) {
  constexpr int PER = F / 4;
  int t = blockIdx.x * blockDim.x + threadIdx.x;
  if (t >= n * PER) return;
  int q = t % PER;
  float4 bb = ((const float4*)b)[q];
  float4 v = ((float4*)agg)[t];
  v.x = fmaxf(v.x + bb.x, 0.0f);
  v.y = fmaxf(v.y + bb.y, 0.0f);
  v.z = fmaxf(v.z + bb.z, 0.0f);
  v.w = fmaxf(v.w + bb.w, 0.0f);
  ((float4*)agg)[t] = v;
}

// out[i] = g[i]*dinv[i]^2 + b2   (self-loop + bias folded into the init; the
// bias add commutes with the subsequent atomic scatter)
__global__ void k_out_init(const float* __restrict__ g,
                           const float* __restrict__ dinv,
                           const float* __restrict__ b2,
                           float* __restrict__ out, int n) {
  int t = blockIdx.x * blockDim.x + threadIdx.x;  // over n*4 float4s (F=16)
  if (t >= n * 4) return;
  int node = t >> 2;
  int q    = t & 3;
  float di = dinv[node];
  float s = di * di;
  float4 v = ((const float4*)g)[t];
  float4 bb = ((const float4*)b2)[q];
  float4 o = {v.x * s + bb.x, v.y * s + bb.y, v.z * s + bb.z, v.w * s + bb.w};
  ((float4*)out)[t] = o;
}

// ---------------------------------------------------------------------------
extern "C" void kernel_launch(void* const* d_in, const int* in_sizes, int n_in,
                              void* d_out, int out_size, void* d_ws,
                              size_t ws_size, hipStream_t stream) {
  constexpr int IN_FEAT = 256, HIDDEN = 64, NCLS = 16;

  const float* x  = (const float*)d_in[0];
  const int*   ei = (const int*)d_in[1];
  const float* W1 = (const float*)d_in[2];
  const float* b1 = (const float*)d_in[3];
  const float* W2 = (const float*)d_in[4];
  const float* b2 = (const float*)d_in[5];
  float* out = (float*)d_out;

  const int N = in_sizes[0] / IN_FEAT;  // 100000
  const int E = in_sizes[1] / 2;        // 1600000
  const int* src = ei;
  const int* dst = ei + E;

  // Workspace layout (floats): dinv[N] | h1[N*64] | agg1[N*64] | g2[N*16]
  float* ws   = (float*)d_ws;
  float* dinv = ws;
  float* h1   = dinv + N;
  float* agg1 = h1 + (size_t)N * HIDDEN;
  float* g2   = agg1 + (size_t)N * HIDDEN;

  const int B = 256;
  const int mtiles  = (N + 15) / 16;
  const int gblocks = (mtiles + 7) / 8;  // 8 waves (16 rows each) per block

  // Normalization: deg -> dinv (in place)
  k_deg_init<<<(N + B - 1) / B, B, 0, stream>>>(dinv, N);
  k_deg_count<<<(E + B - 1) / B, B, 0, stream>>>(dst, dinv, E);
  k_rsqrt_inplace<<<(N + B - 1) / B, B, 0, stream>>>(dinv, N);

  // Layer 1: h1 = x @ W1 on the matrix pipe
  wmma_gemm_f32<IN_FEAT, HIDDEN><<<gblocks, B, 0, stream>>>(x, W1, h1, N);
  k_self_init<HIDDEN><<<((size_t)N * 16 + B - 1) / B, B, 0, stream>>>(h1, dinv, agg1, N);
  k_scatter<HIDDEN><<<((size_t)E * 16 + B - 1) / B, B, 0, stream>>>(src, dst, dinv, h1, agg1, E);
  k_bias_relu<HIDDEN><<<((size_t)N * 16 + B - 1) / B, B, 0, stream>>>(agg1, b1, N);

  // Layer 2: g2 = relu(agg1) @ W2
  wmma_gemm_f32<HIDDEN, NCLS><<<gblocks, B, 0, stream>>>(agg1, W2, g2, N);
  k_out_init<<<((size_t)N * 4 + B - 1) / B, B, 0, stream>>>(g2, dinv, b2, out, N);
  k_scatter<NCLS><<<((size_t)E * 4 + B - 1) / B, B, 0, stream>>>(src, dst, dinv, g2, out, E);
}